// AttnHGCN_44203803410482
// MI455X (gfx1250) — compile-verified
//
#include <hip/hip_runtime.h>
#include <hip/hip_bf16.h>

typedef __attribute__((ext_vector_type(2))) float v2f;
typedef __attribute__((ext_vector_type(8))) float v8f;

#define N_USERS_C 100000
#define N_ENT_C   150000
#define N_REL_C   10
#define CH_C      64
#define N_KG_C    1000000
#define N_UI_C    1000000

// ---- ordered-uint encoding for float atomic max ----
__device__ __forceinline__ unsigned fenc(float f) {
    unsigned u = __float_as_uint(f);
    return (u & 0x80000000u) ? ~u : (u | 0x80000000u);
}
__device__ __forceinline__ float fdec(unsigned e) {
    unsigned u = (e & 0x80000000u) ? (e & 0x7fffffffu) : ~e;
    return __uint_as_float(u);
}

// B[64][16]: cols 0..9 = W_Q @ rel_r ; col 10 = W_UI @ rel_last ; 11..15 = 0
__global__ void build_B_kernel(const float* __restrict__ WQ,
                               const float* __restrict__ WUI,
                               const float* __restrict__ RE,
                               const int*   __restrict__ etype,
                               float* __restrict__ B) {
    int t = blockIdx.x * blockDim.x + threadIdx.x;
    if (t >= 64 * 16) return;
    int k = t >> 4, r = t & 15;
    float acc = 0.f;
    if (r < N_REL_C) {
        const float* w  = WQ + k * CH_C;
        const float* re = RE + r * CH_C;
        #pragma unroll 8
        for (int c = 0; c < CH_C; ++c) acc += w[c] * re[c];
    } else if (r == N_REL_C) {
        int rl = etype[N_KG_C - 1] - 1;
        const float* w  = WUI + k * CH_C;
        const float* re = RE + rl * CH_C;
        #pragma unroll 8
        for (int c = 0; c < CH_C; ++c) acc += w[c] * re[c];
    }
    B[k * 16 + r] = acc;
}

// S[M,16] = X[M,64] @ B[64,16] via V_WMMA_F32_16X16X4_F32; one 16x16 tile per wave32.
__global__ void score_gemm_wmma(const float* __restrict__ X,
                                const float* __restrict__ B,
                                float* __restrict__ S, int M) {
    int lane = threadIdx.x & 31;
    int wave = threadIdx.x >> 5;
    int tile = blockIdx.x * 8 + wave;
    int m0 = tile * 16;
    if (m0 >= M) return;                 // uniform per wave -> EXEC all-ones for WMMA
    int half = lane >> 4;                // selects K pair {0,1} vs {2,3} within k0..k0+3
    int nr = lane & 15;                  // A row within tile == B/D column

    int mrow = m0 + nr;
    if (mrow >= M) mrow = M - 1;         // clamp (results discarded on store)
    const float* xr = X + (size_t)mrow * CH_C;

    v8f acc = {};
    #pragma unroll
    for (int k0 = 0; k0 < CH_C; k0 += 4) {
        int kk = k0 + half * 2;
        v2f a; a.x = xr[kk];            a.y = xr[kk + 1];
        v2f b; b.x = B[kk * 16 + nr];   b.y = B[(kk + 1) * 16 + nr];
        acc = __builtin_amdgcn_wmma_f32_16x16x4_f32(
            false, a, false, b, (short)0, acc, false, false);
    }
    #pragma unroll
    for (int j = 0; j < 8; ++j) {
        int row = m0 + j + half * 8;     // C/D layout: VGPR j -> M = j (+8 for hi lanes)
        if (row < M) S[(size_t)row * 16 + nr] = acc[j];
    }
}

// s_user[u] = user_emb[u] . B[:,10]   (one wave per user, float2 per lane)
__global__ void user_score_kernel(const float* __restrict__ U,
                                  const float* __restrict__ B,
                                  float* __restrict__ su) {
    int lane = threadIdx.x & 31;
    int u = (blockIdx.x * blockDim.x + threadIdx.x) >> 5;
    if (u >= N_USERS_C) return;
    const float* ur = U + (size_t)u * CH_C;
    int c = lane * 2;
    float s = ur[c] * B[c * 16 + 10] + ur[c + 1] * B[(c + 1) * 16 + 10];
    #pragma unroll
    for (int off = 16; off; off >>= 1) s += __shfl_xor(s, off, 32);
    if (lane == 0) su[u] = s;
}

// KG pass 1: score + segment max (over head)
__global__ void kg_score_max(const int* __restrict__ EI, const int* __restrict__ ET,
                             const float* __restrict__ S, float* __restrict__ esc,
                             unsigned* __restrict__ emax) {
    int e = blockIdx.x * blockDim.x + threadIdx.x;
    if (e >= N_KG_C) return;
    int h = EI[e], t = EI[N_KG_C + e], r = ET[e] - 1;
    float sc = S[(size_t)h * 16 + r] + S[(size_t)t * 16 + r];
    esc[e] = sc;
    atomicMax(&emax[h], fenc(sc));
}

// KG pass 2: exp-sum
__global__ void kg_expsum(const int* __restrict__ EI, const float* __restrict__ esc,
                          const unsigned* __restrict__ emax, float* __restrict__ esum) {
    int e = blockIdx.x * blockDim.x + threadIdx.x;
    if (e >= N_KG_C) return;
    int h = EI[e];
    atomicAdd(&esum[h], expf(esc[e] - fdec(emax[h])));
}

// KG pass 3: scatter attn * entity[tail] * rel  (one wave per edge)
__global__ void kg_scatter(const int* __restrict__ EI, const int* __restrict__ ET,
                           const float* __restrict__ esc, const unsigned* __restrict__ emax,
                           const float* __restrict__ esum, const float* __restrict__ E,
                           const float* __restrict__ RE, float* __restrict__ agg) {
    int lane = threadIdx.x & 31;
    int e = (blockIdx.x * blockDim.x + threadIdx.x) >> 5;
    if (e >= N_KG_C) return;
    int h = EI[e], t = EI[N_KG_C + e], r = ET[e] - 1;
    float attn = expf(esc[e] - fdec(emax[h])) / (esum[h] + 1e-16f);
    int c = lane * 2;
    float2 v  = *(const float2*)(E  + (size_t)t * CH_C + c);
    float2 rl = *(const float2*)(RE + (size_t)r * CH_C + c);
    atomicAdd(&agg[(size_t)h * CH_C + c],     attn * v.x * rl.x);
    atomicAdd(&agg[(size_t)h * CH_C + c + 1], attn * v.y * rl.y);
}

// UI pass 1: score + segment max (over user)
__global__ void ui_score_max(const int* __restrict__ UIE, const float* __restrict__ su,
                             const float* __restrict__ S, float* __restrict__ esc,
                             unsigned* __restrict__ umax) {
    int e = blockIdx.x * blockDim.x + threadIdx.x;
    if (e >= N_UI_C) return;
    int u = UIE[e], i = UIE[N_UI_C + e];
    float sc = su[u] + S[(size_t)i * 16 + 10];
    esc[e] = sc;
    atomicMax(&umax[u], fenc(sc));
}

__global__ void ui_expsum(const int* __restrict__ UIE, const float* __restrict__ esc,
                          const unsigned* __restrict__ umax, float* __restrict__ usum) {
    int e = blockIdx.x * blockDim.x + threadIdx.x;
    if (e >= N_UI_C) return;
    int u = UIE[e];
    atomicAdd(&usum[u], expf(esc[e] - fdec(umax[u])));
}

// UI pass 3: scatter w_e * attn * entity[i]   (one wave per edge)
__global__ void ui_scatter(const int* __restrict__ UIE, const float* __restrict__ IW,
                           const float* __restrict__ esc, const unsigned* __restrict__ umax,
                           const float* __restrict__ usum, const float* __restrict__ E,
                           float* __restrict__ agg) {
    int lane = threadIdx.x & 31;
    int e = (blockIdx.x * blockDim.x + threadIdx.x) >> 5;
    if (e >= N_UI_C) return;
    int u = UIE[e], i = UIE[N_UI_C + e];
    float w = IW[e] * expf(esc[e] - fdec(umax[u])) / (usum[u] + 1e-16f);
    int c = lane * 2;
    float2 v = *(const float2*)(E + (size_t)i * CH_C + c);
    atomicAdd(&agg[(size_t)u * CH_C + c],     w * v.x);
    atomicAdd(&agg[(size_t)u * CH_C + c + 1], w * v.y);
}

// L2-normalize row in place (becomes next hop's input) + accumulate into residual.
__global__ void norm_acc(float* __restrict__ agg, float* __restrict__ res, int nrows) {
    int lane = threadIdx.x & 31;
    int row = (blockIdx.x * blockDim.x + threadIdx.x) >> 5;
    if (row >= nrows) return;
    int c = lane * 2;
    float2 x = *(float2*)(agg + (size_t)row * CH_C + c);
    float s = x.x * x.x + x.y * x.y;
    #pragma unroll
    for (int off = 16; off; off >>= 1) s += __shfl_xor(s, off, 32);
    float inv = 1.f / fmaxf(sqrtf(s), 1e-12f);
    x.x *= inv; x.y *= inv;
    *(float2*)(agg + (size_t)row * CH_C + c) = x;
    float2* rp = (float2*)(res + (size_t)row * CH_C + c);
    float2 rv = *rp;
    rv.x += x.x; rv.y += x.y;
    *rp = rv;
}

__global__ void copy_f4(const float4* __restrict__ src, float4* __restrict__ dst, int n4) {
    int i = blockIdx.x * blockDim.x + threadIdx.x;
    if (i < n4) dst[i] = src[i];
}

extern "C" void kernel_launch(void* const* d_in, const int* in_sizes, int n_in,
                              void* d_out, int out_size, void* d_ws, size_t ws_size,
                              hipStream_t stream) {
    const float* user_in = (const float*)d_in[0];
    const float* ent_in  = (const float*)d_in[1];
    const int*   EI      = (const int*)d_in[2];   // [2, N_KG]
    const int*   ET      = (const int*)d_in[3];
    const int*   UIE     = (const int*)d_in[4];   // [2, N_UI]
    const float* IW      = (const float*)d_in[5];
    const float* RE      = (const float*)d_in[6];
    const float* WQ      = (const float*)d_in[7];
    const float* WUI     = (const float*)d_in[8];

    float* res_ent = (float*)d_out;
    float* res_usr = res_ent + (size_t)N_ENT_C * CH_C;

    // workspace carve-out (256-byte aligned chunks)
    float* ws = (float*)d_ws;
    size_t off = 0;
    auto alloc = [&](size_t n) { float* p = ws + off; off += (n + 63) & ~(size_t)63; return p; };
    float*    B    = alloc(64 * 16);
    float*    Sent = alloc((size_t)N_ENT_C * 16);
    float*    su   = alloc(N_USERS_C);
    float*    esc  = alloc(N_KG_C);              // shared by KG then UI passes
    unsigned* emax = (unsigned*)alloc(N_ENT_C);
    float*    esum = alloc(N_ENT_C);
    unsigned* umax = (unsigned*)alloc(N_USERS_C);
    float*    usum = alloc(N_USERS_C);
    float*    entBuf[2] = { alloc((size_t)N_ENT_C * CH_C),   alloc((size_t)N_ENT_C * CH_C) };
    float*    usrBuf[2] = { alloc((size_t)N_USERS_C * CH_C), alloc((size_t)N_USERS_C * CH_C) };

    // residual init: res = input embeddings
    copy_f4<<<(N_ENT_C * CH_C / 4 + 255) / 256, 256, 0, stream>>>(
        (const float4*)ent_in, (float4*)res_ent, N_ENT_C * CH_C / 4);
    copy_f4<<<(N_USERS_C * CH_C / 4 + 255) / 256, 256, 0, stream>>>(
        (const float4*)user_in, (float4*)res_usr, N_USERS_C * CH_C / 4);

    // hop-invariant projection matrix
    build_B_kernel<<<4, 256, 0, stream>>>(WQ, WUI, RE, ET, B);

    const float* curE = ent_in;
    const float* curU = user_in;
    const int ntilesE = (N_ENT_C + 15) / 16;

    for (int hop = 0; hop < 2; ++hop) {
        float* aggE = entBuf[hop];
        float* aggU = usrBuf[hop];
        hipMemsetAsync(aggE, 0, (size_t)N_ENT_C * CH_C * sizeof(float), stream);
        hipMemsetAsync(aggU, 0, (size_t)N_USERS_C * CH_C * sizeof(float), stream);
        hipMemsetAsync(emax, 0, N_ENT_C * sizeof(unsigned), stream);
        hipMemsetAsync(esum, 0, N_ENT_C * sizeof(float), stream);
        hipMemsetAsync(umax, 0, N_USERS_C * sizeof(unsigned), stream);
        hipMemsetAsync(usum, 0, N_USERS_C * sizeof(float), stream);

        // score projections
        score_gemm_wmma<<<(ntilesE + 7) / 8, 256, 0, stream>>>(curE, B, Sent, N_ENT_C);
        user_score_kernel<<<(N_USERS_C * 32 + 255) / 256, 256, 0, stream>>>(curU, B, su);

        // KG attention + aggregation
        kg_score_max<<<(N_KG_C + 255) / 256, 256, 0, stream>>>(EI, ET, Sent, esc, emax);
        kg_expsum<<<(N_KG_C + 255) / 256, 256, 0, stream>>>(EI, esc, emax, esum);
        kg_scatter<<<(N_KG_C + 7) / 8, 256, 0, stream>>>(EI, ET, esc, emax, esum, curE, RE, aggE);

        // UI attention + aggregation (reuses esc after KG passes complete on stream)
        ui_score_max<<<(N_UI_C + 255) / 256, 256, 0, stream>>>(UIE, su, Sent, esc, umax);
        ui_expsum<<<(N_UI_C + 255) / 256, 256, 0, stream>>>(UIE, esc, umax, usum);
        ui_scatter<<<(N_UI_C + 7) / 8, 256, 0, stream>>>(UIE, IW, esc, umax, usum, curE, aggU);

        // normalize in place + accumulate residual
        norm_acc<<<(N_ENT_C + 7) / 8, 256, 0, stream>>>(aggE, res_ent, N_ENT_C);
        norm_acc<<<(N_USERS_C + 7) / 8, 256, 0, stream>>>(aggU, res_usr, N_USERS_C);

        curE = aggE;
        curU = aggU;
    }
}